// smoothTransformer2D_38354057954072
// MI455X (gfx1250) — compile-verified
//
#include <hip/hip_runtime.h>
#include <math.h>

// MI455X / gfx1250, wave32.
typedef __attribute__((ext_vector_type(2))) float v2f;
typedef __attribute__((ext_vector_type(8))) float v8f;

#define BB 16
#define HH 512
#define WW 512
#define CC 4

__device__ __forceinline__ float logistic2(float x) {
    // MAXGRAD = 2:  c / (1 + (c-1)*exp(-x)) = 2 / (1 + exp(-x))
    return 2.0f / (1.0f + __expf(-x));
}

// ---------------------------------------------------------------------------
// Kernel 1: per-row inclusive cumsum of logistic(defgrad[...,0]) -> g3[...,0]
// One wave32 per row; wave-shuffle Hillis-Steele scan, carried across 16
// chunks of 32. No barriers, no LDS.
// ---------------------------------------------------------------------------
__global__ __launch_bounds__(256) void k_row_scan(const float* __restrict__ dg,
                                                  float* __restrict__ g3) {
    const int lane = threadIdx.x & 31;
    const int row  = blockIdx.x * 8 + (threadIdx.x >> 5);   // row = b*H + h
    const float* src = dg + (size_t)row * WW * 2;           // comp 0
    float* dst = g3 + (size_t)row * WW * 3;                 // comp 0 slot
    float carry = 0.0f;
    for (int c = 0; c < WW; c += 32) {
        float s = logistic2(src[(size_t)(c + lane) * 2]);
#pragma unroll
        for (int d = 1; d < 32; d <<= 1) {
            float t = __shfl_up(s, d, 32);
            if (lane >= d) s += t;
        }
        s += carry;
        dst[(size_t)(c + lane) * 3] = s;
        carry = __shfl(s, 31, 32);
    }
}

// ---------------------------------------------------------------------------
// Kernel 2: per-column inclusive cumsum of logistic(defgrad[...,1]) -> g3[...,1]
// One thread per (b,w) column; unroll-4 for ILP on the serial add chain;
// global_prefetch_b8 covers the 4KB-stride walk down the column.
// ---------------------------------------------------------------------------
__global__ __launch_bounds__(256) void k_col_scan(const float* __restrict__ dg,
                                                  float* __restrict__ g3) {
    const int t = blockIdx.x * blockDim.x + threadIdx.x;    // 0 .. B*W-1
    const int b = t / WW;
    const int w = t % WW;
    const float* src = dg + (((size_t)b * HH) * WW + w) * 2 + 1;
    float* dst = g3 + (((size_t)b * HH) * WW + w) * 3 + 1;
    const size_t sstep = (size_t)WW * 2;
    const size_t dstep = (size_t)WW * 3;
    float acc = 0.0f;
    for (int h = 0; h < HH; h += 4) {
        const int hp = (h + 16 < HH) ? (h + 16) : (HH - 1);
        __builtin_prefetch(src + (size_t)hp * sstep, 0, 1); // global_prefetch_b8
        float l0 = logistic2(src[(size_t)(h + 0) * sstep]);
        float l1 = logistic2(src[(size_t)(h + 1) * sstep]);
        float l2 = logistic2(src[(size_t)(h + 2) * sstep]);
        float l3 = logistic2(src[(size_t)(h + 3) * sstep]);
        acc += l0; dst[(size_t)(h + 0) * dstep] = acc;
        acc += l1; dst[(size_t)(h + 1) * dstep] = acc;
        acc += l2; dst[(size_t)(h + 2) * dstep] = acc;
        acc += l3; dst[(size_t)(h + 3) * dstep] = acc;
    }
}

// ---------------------------------------------------------------------------
// Kernel 3: fused residual-affine transform (via V_WMMA_F32_16X16X4_F32) +
// bilinear gather. One thread per pixel; each wave transforms its 32 pixels
// with two WMMAs: A (16x4) = [x, y, 1, 0] rows, B (4x16) = Aaff^T.
// D (16x16) holds grid3 components in columns 0..2; scattered through LDS.
// ---------------------------------------------------------------------------
__global__ __launch_bounds__(256) void k_warp_sample(const float* __restrict__ im,
                                                     const float* __restrict__ affine,
                                                     float* __restrict__ out,
                                                     float* __restrict__ g3) {
    __shared__ float xch[8 * 96];                           // 8 waves * 32 px * 3
    const int tid  = blockIdx.x * 256 + threadIdx.x;        // pixel id
    const int lane = threadIdx.x & 31;
    const int n    = lane & 15;                             // column index N
    const bool hi  = lane >= 16;
    const int ldsBase = (threadIdx.x >> 5) * 96;
    const int b = tid / (HH * WW);                          // uniform per block

    // raw cumsums staged in the grid3 output region by kernels 1/2
    float xr = g3[(size_t)tid * 3 + 0];
    float yr = g3[(size_t)tid * 3 + 1];

    // B-matrix: B[k][n] = Aaff[n][k], Aaff = affine(b) + I, zero-padded n>=3.
    // Layout: VGPR0 = {lanes0-15: K=0, lanes16-31: K=2}, VGPR1 = {K=1, K=3}.
    float b0 = 0.0f, b1 = 0.0f;
    if (n < 3) {
        const float* ab = affine + b * 9 + n * 3;
        float an0 = ab[0] + (n == 0 ? 1.0f : 0.0f);
        float an1 = ab[1] + (n == 1 ? 1.0f : 0.0f);
        float an2 = ab[2] + (n == 2 ? 1.0f : 0.0f);
        b0 = hi ? an2 : an0;
        b1 = hi ? 0.0f : an1;
    }
    v2f Bm = {b0, b1};

#pragma unroll
    for (int g = 0; g < 2; ++g) {
        // A-matrix: lanes0-15 row M holds {K0=x, K1=y}; lanes16-31 {K2=1, K3=0}
        float xA = __shfl(xr, g * 16 + n, 32);
        float yA = __shfl(yr, g * 16 + n, 32);
        v2f Am = {hi ? 1.0f : xA, hi ? 0.0f : yA};
        v8f Cz = {0.f, 0.f, 0.f, 0.f, 0.f, 0.f, 0.f, 0.f};
        v8f D = __builtin_amdgcn_wmma_f32_16x16x4_f32(
            /*neg_a=*/false, Am, /*neg_b=*/false, Bm,
            /*c_mod=*/(short)0, Cz, /*reuse_a=*/false, /*reuse_b=*/false);
        // D layout: VGPR r, lanes0-15 -> (M=r, N=lane); lanes16-31 -> (M=r+8)
        if (n < 3) {
            const int mbase = g * 16 + (hi ? 8 : 0);
#pragma unroll
            for (int r = 0; r < 8; ++r)
                xch[ldsBase + (mbase + r) * 3 + n] = D[r];
        }
    }
    __syncthreads();

    float gx = xch[ldsBase + lane * 3 + 0];
    float gy = xch[ldsBase + lane * 3 + 1];
    float gz = xch[ldsBase + lane * 3 + 2];

    g3[(size_t)tid * 3 + 0] = gx;
    g3[(size_t)tid * 3 + 1] = gy;
    g3[(size_t)tid * 3 + 2] = gz;

    // bilinear sampling, clamped corners (matches reference exactly)
    float fx = floorf(gx), fy = floorf(gy);
    int x0 = min(max((int)fx, 0), WW - 1);
    int x1 = min(max((int)fx + 1, 0), WW - 1);
    int y0 = min(max((int)fy, 0), HH - 1);
    int y1 = min(max((int)fy + 1, 0), HH - 1);
    float x0f = (float)x0, x1f = (float)x1, y0f = (float)y0, y1f = (float)y1;
    float wa = (x1f - gx) * (y1f - gy);
    float wb = (x1f - gx) * (gy - y0f);
    float wc = (gx - x0f) * (y1f - gy);
    float wd = (gx - x0f) * (gy - y0f);

    const float4* imv = (const float4*)im;                  // C=4 -> b128 loads
    const size_t base = (size_t)b * HH * WW;
    float4 Ia = imv[base + (size_t)y0 * WW + x0];
    float4 Ib = imv[base + (size_t)y1 * WW + x0];
    float4 Ic = imv[base + (size_t)y0 * WW + x1];
    float4 Id = imv[base + (size_t)y1 * WW + x1];

    float4 o;
    o.x = wa * Ia.x + wb * Ib.x + wc * Ic.x + wd * Id.x;
    o.y = wa * Ia.y + wb * Ib.y + wc * Ic.y + wd * Id.y;
    o.z = wa * Ia.z + wb * Ib.z + wc * Ic.z + wd * Id.z;
    o.w = wa * Ia.w + wb * Ib.w + wc * Ic.w + wd * Id.w;
    ((float4*)out)[tid] = o;                                // b128 store
}

// ---------------------------------------------------------------------------
extern "C" void kernel_launch(void* const* d_in, const int* in_sizes, int n_in,
                              void* d_out, int out_size, void* d_ws, size_t ws_size,
                              hipStream_t stream) {
    (void)in_sizes; (void)n_in; (void)d_ws; (void)ws_size; (void)out_size;
    const float* im = (const float*)d_in[0];
    const float* dg = (const float*)d_in[1];
    const float* af = (const float*)d_in[2];
    float* out = (float*)d_out;
    float* g3  = out + (size_t)BB * HH * WW * CC;   // second output region

    k_row_scan <<<(BB * HH) / 8,        256, 0, stream>>>(dg, g3);
    k_col_scan <<<(BB * WW) / 256,      256, 0, stream>>>(dg, g3);
    k_warp_sample<<<(BB * HH * WW) / 256, 256, 0, stream>>>(im, af, out, g3);
}